// TriAttnModel_19980187861851
// MI455X (gfx1250) — compile-verified
//
#include <hip/hip_runtime.h>
#include <hip/hip_bf16.h>
#include <math.h>

// TriAttn for B=16, S=2048, D=32 on gfx1250 (MI455X).
// Strategy: flash-style column-softmax attention built on V_WMMA_F32_16X16X4_F32.
//  - proj_kernel: F_av/F_al/F_vl = relu(cat(x1,x2) @ w.T + b)  -> workspace
//  - pass1_stats: per-column online max/sum of exp(C[:,t]) where C = F1 @ F23^T
//  - pass2_attn : recompute C tiles, P = exp(C-M)/Z, T = P @ F1, out = T * F23

typedef __attribute__((ext_vector_type(2))) float v2f;
typedef __attribute__((ext_vector_type(8))) float v8f;

#define WMMA4(A, Bm, C) \
  __builtin_amdgcn_wmma_f32_16x16x4_f32(false, (A), false, (Bm), (short)0, (C), false, false)

static constexpr int kB = 16;
static constexpr int kS = 2048;
static constexpr int kD = 32;
static constexpr size_t kTensorElems = (size_t)kB * kS * kD;   // per projected tensor
static constexpr size_t kStatsBase   = 3 * kTensorElems;       // floats
static constexpr size_t kStatsSz     = (size_t)3 * kB * kS;    // per stat array

// ---------------------------------------------------------------------------
// Projection: y[b,s,d] = relu( sum_k x1[k]*w[d,k] + x2[k]*w[d,32+k] + bias[d] )
// grid (S/8, B, 3), block 256 (8 rows x 32 outputs)
// ---------------------------------------------------------------------------
__global__ __launch_bounds__(256) void proj_kernel(
    const float* __restrict__ Fa, const float* __restrict__ Fv, const float* __restrict__ Fl,
    const float* __restrict__ w1, const float* __restrict__ b1,
    const float* __restrict__ w2, const float* __restrict__ b2,
    const float* __restrict__ w3, const float* __restrict__ b3,
    float* __restrict__ ws) {
  __shared__ float wl[32 * 65];   // padded: bank = (65d+k)%64 = (d+k)%64
  __shared__ float bl[32];
  __shared__ float xs[8 * 64];

  const int a = blockIdx.z;
  const int b = blockIdx.y;
  const float* x1   = (a == 2) ? Fv : Fa;
  const float* x2   = (a == 0) ? Fv : Fl;
  const float* w    = (a == 0) ? w1 : (a == 1) ? w2 : w3;
  const float* bias = (a == 0) ? b1 : (a == 1) ? b2 : b3;

  const int tid = threadIdx.x;
  for (int i = tid; i < 32 * 64; i += 256) wl[(i >> 6) * 65 + (i & 63)] = w[i];
  if (tid < 32) bl[tid] = bias[tid];

  const int row0 = blockIdx.x * 8;
  for (int i = tid; i < 8 * 64; i += 256) {
    const int r = i >> 6, k = i & 63;
    const size_t g = ((size_t)b * kS + row0 + r) * kD;
    xs[i] = (k < 32) ? x1[g + k] : x2[g + (k - 32)];
  }
  __syncthreads();

  const int d = tid & 31, r = tid >> 5;
  float y = bl[d];
#pragma unroll 8
  for (int k = 0; k < 64; ++k) y += xs[r * 64 + k] * wl[d * 65 + k];
  y = fmaxf(y, 0.0f);
  ws[(size_t)a * kTensorElems + ((size_t)b * kS + row0 + r) * kD + d] = y;
}

// ---------------------------------------------------------------------------
// Pass 1: for each column t, M[t] = max_s C[s,t], Z[t] = sum_s exp(C[s,t]-M[t])
// C[s,t] = F1[s] . F23[t].  One wave per 16-column strip, online softmax.
// grid (S/16, B, 3), block 32.
// ---------------------------------------------------------------------------
__global__ __launch_bounds__(32) void pass1_stats(
    const float* __restrict__ Fa, const float* __restrict__ Fv,
    const float* __restrict__ Fl, float* __restrict__ ws) {
  const int tstrip = blockIdx.x, b = blockIdx.y, a = blockIdx.z;
  const float* F23 = ws + (size_t)a * kTensorElems;
  const float* F1  = (a == 0) ? Fl : (a == 1) ? Fv : Fa;

  const int lane = threadIdx.x & 31;
  const int col  = lane & 15;
  const int hi   = lane >> 4;
  const int t0   = tstrip * 16;

  // B operand (fixed per strip): B[k,n] = F23[t0+n, kblk*4 + k]
  const float* f23row = F23 + ((size_t)b * kS + t0 + col) * kD + 2 * hi;
  v2f bb[8];
#pragma unroll
  for (int kk = 0; kk < 8; ++kk) bb[kk] = *(const v2f*)(f23row + 4 * kk);

  const float* f1base = F1 + ((size_t)b * kS + col) * kD + 2 * hi;

  float m = -INFINITY, z = 0.0f;
  for (int st = 0; st < kS / 16; ++st) {
    const float* f1row = f1base + (size_t)st * 16 * kD;
    v8f c = {};
#pragma unroll
    for (int kk = 0; kk < 8; ++kk) {
      v2f aa = *(const v2f*)(f1row + 4 * kk);
      c = WMMA4(aa, bb[kk], c);
    }
    // column-wise stats: each lane holds rows {g+8*hi} of column (t0+col)
    float tmax = c[0];
#pragma unroll
    for (int g = 1; g < 8; ++g) tmax = fmaxf(tmax, c[g]);
    tmax = fmaxf(tmax, __shfl_xor(tmax, 16, 32));
    const float newm = fmaxf(m, tmax);
    float ps = 0.0f;
#pragma unroll
    for (int g = 0; g < 8; ++g) ps += __expf(c[g] - newm);
    ps += __shfl_xor(ps, 16, 32);
    z = z * __expf(m - newm) + ps;
    m = newm;
  }
  if (hi == 0) {
    float* Mb = ws + kStatsBase;
    float* Zb = Mb + kStatsSz;
    const size_t idx = ((size_t)a * kB + b) * kS + t0 + col;
    Mb[idx] = m;
    Zb[idx] = z;
  }
}

// ---------------------------------------------------------------------------
// Pass 2: T[s,:] = sum_t exp(C[s,t]-M[t])/Z[t] * F1[t,:]; out = T * F23.
// One wave per 16-row s-strip; loops over t tiles; P transposed via LDS.
// grid (S/16, B, 3), block 32.
// ---------------------------------------------------------------------------
__global__ __launch_bounds__(32) void pass2_attn(
    const float* __restrict__ Fa, const float* __restrict__ Fv,
    const float* __restrict__ Fl, const float* __restrict__ ws,
    float* __restrict__ out) {
  __shared__ float plds[16 * 20];   // [s_local][t_local], row stride 20 (pad)

  const int sstrip = blockIdx.x, b = blockIdx.y, a = blockIdx.z;
  const float* F23 = ws + (size_t)a * kTensorElems;
  const float* F1  = (a == 0) ? Fl : (a == 1) ? Fv : Fa;
  const float* Mb  = ws + kStatsBase + ((size_t)a * kB + b) * kS;
  const float* Zb  = Mb + kStatsSz;

  const int lane = threadIdx.x & 31;
  const int col  = lane & 15;
  const int hi   = lane >> 4;
  const int s0   = sstrip * 16;

  // A operand of the score GEMM (fixed): A[m,k] = F1[s0+m, kblk*4+k]
  const float* f1s = F1 + ((size_t)b * kS + s0 + col) * kD + 2 * hi;
  v2f fa8[8];
#pragma unroll
  for (int kk = 0; kk < 8; ++kk) fa8[kk] = *(const v2f*)(f1s + 4 * kk);

  const float* f23b = F23 + (size_t)b * kS * kD;
  const float* f1b  = F1 + (size_t)b * kS * kD;

  v8f acc0 = {};  // T[:, 0..15]
  v8f acc1 = {};  // T[:, 16..31]

  for (int tt = 0; tt < kS / 16; ++tt) {
    const int t0 = tt * 16;
    // scores C[s0+.., t0+..]
    const float* f23row = f23b + (size_t)(t0 + col) * kD + 2 * hi;
    v8f c = {};
#pragma unroll
    for (int kk = 0; kk < 8; ++kk) {
      v2f bbv = *(const v2f*)(f23row + 4 * kk);
      c = WMMA4(fa8[kk], bbv, c);
    }
    const float mt   = Mb[t0 + col];
    const float zinv = 1.0f / Zb[t0 + col];
    // c[g] = C[s0 + g + 8*hi, t0 + col]  ->  P in [s_local][t_local] LDS layout
#pragma unroll
    for (int g = 0; g < 8; ++g) {
      plds[(g + 8 * hi) * 20 + col] = __expf(c[g] - mt) * zinv;
    }
    // T += P @ F1[t0:t0+16, :]  (K = 16 -> 4 WMMAs per d-chunk)
#pragma unroll
    for (int kk = 0; kk < 4; ++kk) {
      v2f a2 = *(const v2f*)&plds[col * 20 + 4 * kk + 2 * hi];
      const int trow = t0 + 4 * kk + 2 * hi;
      v2f b0, b1v;
      b0[0]  = f1b[(size_t)trow * kD + col];
      b0[1]  = f1b[(size_t)(trow + 1) * kD + col];
      b1v[0] = f1b[(size_t)trow * kD + 16 + col];
      b1v[1] = f1b[(size_t)(trow + 1) * kD + 16 + col];
      acc0 = WMMA4(a2, b0, acc0);
      acc1 = WMMA4(a2, b1v, acc1);
    }
  }

  // Epilogue: out[b, s, a*32 + d] = T[s,d] * F23[b,s,d]
#pragma unroll
  for (int g = 0; g < 8; ++g) {
    const int srow = s0 + g + 8 * hi;
    const size_t rb = (size_t)b * kS + srow;
    const float f0 = F23[rb * kD + col];
    const float f1v = F23[rb * kD + 16 + col];
    out[rb * 96 + a * 32 + col]      = acc0[g] * f0;
    out[rb * 96 + a * 32 + 16 + col] = acc1[g] * f1v;
  }
}

// ---------------------------------------------------------------------------
extern "C" void kernel_launch(void* const* d_in, const int* in_sizes, int n_in,
                              void* d_out, int out_size, void* d_ws, size_t ws_size,
                              hipStream_t stream) {
  const float* Fa = (const float*)d_in[0];
  const float* Fv = (const float*)d_in[1];
  const float* Fl = (const float*)d_in[2];
  const float* w1 = (const float*)d_in[3];
  const float* b1 = (const float*)d_in[4];
  const float* w2 = (const float*)d_in[5];
  const float* b2 = (const float*)d_in[6];
  const float* w3 = (const float*)d_in[7];
  const float* b3 = (const float*)d_in[8];
  float* ws  = (float*)d_ws;
  float* out = (float*)d_out;

  dim3 gproj(kS / 8, kB, 3);
  proj_kernel<<<gproj, 256, 0, stream>>>(Fa, Fv, Fl, w1, b1, w2, b2, w3, b3, ws);

  dim3 gattn(kS / 16, kB, 3);
  pass1_stats<<<gattn, 32, 0, stream>>>(Fa, Fv, Fl, ws);
  pass2_attn<<<gattn, 32, 0, stream>>>(Fa, Fv, Fl, ws, out);
}